// LRN_19421842113133
// MI455X (gfx1250) — compile-verified
//
#include <hip/hip_runtime.h>
#include <cstdint>

// LRN over channel axis, NHWC, ch=256, window n=256, alpha/n=1, k=0, beta=0.5.
// win[c] = P[min(c+127,255)] - (c>=129 ? P[c-129] : 0),  P = inclusive prefix of x^2.
// out = x * rsqrt(win).
//
// One wave32 per pixel. Input staged via CDNA5 async global->LDS loads
// (ASYNCcnt path). Prefix built with a wave32 shuffle scan, then written into
// a zero/total-padded LDS row so the window gather is 4 aligned ds_load_b128
// with zero clamp arithmetic:
//   row[0..132]   = 0
//   row[133 + c]  = P[c]          (c = 0..255)
//   row[388..515] = total         (row[388] == P[255] == total, benign overlap)
//   lo[c] = row[c + 4],  up[c] = row[c + 260],  win = up - lo.

typedef float f32x4 __attribute__((ext_vector_type(4)));  // native vector (NT-store OK)

#define CH 256
#define WPB 8              // waves (pixels) per block
#define THREADS (WPB * 32)
#define ROW 516            // padded prefix row length (stride: 516*4 B, 16B multiple)
#define PAD 133            // P[c] lives at row[PAD + c]; PAD % 4 == 1 aligns both gathers

__global__ __launch_bounds__(THREADS) void lrn_kernel(const float* __restrict__ x,
                                                      float* __restrict__ out,
                                                      int npix) {
  __shared__ __align__(16) float s_x[WPB][CH];  // staged input (1 KB per wave)
  __shared__ __align__(16) float s_r[WPB][ROW]; // padded prefix row (~2 KB per wave)

  const int wave = (int)(threadIdx.x >> 5);
  const int lane = (int)(threadIdx.x & 31);
  const int pix = (int)blockIdx.x * WPB + wave;   // 32-bit: npix = 100352
  if (pix >= npix) return;

  const size_t elt0 = (size_t)pix << 8;           // pix * CH
  const float* __restrict__ src = x + elt0;

  // ---- CDNA5 async global->LDS staging ------------------------------------
  // Two B128 async loads per wave: 32 lanes * 16B = 512B each -> 1 KB total.
  // GVS addressing: mem = SGPR64 + VGPR32 + IOFFSET; LDS dst = VGPR + IOFFSET.
  // Generic pointers to LDS carry the workgroup-relative LDS offset in the
  // low 32 bits (flat->LDS aperture truncation).
  {
    unsigned lds_dst = (unsigned)(uintptr_t)(&s_x[wave][0]) + (unsigned)(lane * 16);
    unsigned voff = (unsigned)(lane * 16);
    unsigned long long gbase = (unsigned long long)(uintptr_t)src;
    asm volatile(
        "global_load_async_to_lds_b128 %0, %1, %2 offset:0\n\t"
        "global_load_async_to_lds_b128 %0, %1, %2 offset:512"
        :
        : "v"(lds_dst), "v"(voff), "s"(gbase)
        : "memory");
  }

  // ---- zero-fill the low pad while the async loads are in flight ----------
  const f32x4 z4 = {0.0f, 0.0f, 0.0f, 0.0f};
  *(f32x4*)(&s_r[wave][lane * 4]) = z4;           // rows 0..127
  if (lane == 0) *(f32x4*)(&s_r[wave][128]) = z4; // rows 128..131
  if (lane == 1) s_r[wave][132] = 0.0f;           // row 132

  asm volatile("s_wait_asynccnt 0" ::: "memory");

  // ---- per-lane: 8 contiguous channels ------------------------------------
  const int c0 = lane * 8;
  f32x4 a = *(const f32x4*)(&s_x[wave][c0]);
  f32x4 b = *(const f32x4*)(&s_x[wave][c0 + 4]);

  // squares + lane-local inclusive prefix
  float p0 = a.x * a.x;
  float p1 = p0 + a.y * a.y;
  float p2 = p1 + a.z * a.z;
  float p3 = p2 + a.w * a.w;
  float p4 = p3 + b.x * b.x;
  float p5 = p4 + b.y * b.y;
  float p6 = p5 + b.z * b.z;
  float p7 = p6 + b.w * b.w;
  const float lsum = p7;

  // ---- wave32 inclusive scan of per-lane sums (Hillis-Steele) -------------
  float v = lsum;
#pragma unroll
  for (int d = 1; d < 32; d <<= 1) {
    float t = __shfl_up(v, d, 32);
    if (lane >= d) v += t;
  }
  const float excl = v - lsum;  // exclusive prefix for this lane

  // inclusive per-channel prefix values
  const float q0 = excl + p0, q1 = excl + p1, q2 = excl + p2, q3 = excl + p3;
  const float q4 = excl + p4, q5 = excl + p5, q6 = excl + p6, q7 = excl + p7;

  // total broadcast bit-identical to lane 31's q7 so the row[388] overlap
  // between the P store and the totals fill is a benign same-value race.
  const float total = __shfl(q7, 31, 32);

  // P stores (scalar; base PAD=133 is deliberately vector-misaligned so that
  // both gather runs below are vector-ALIGNED)
  float* __restrict__ rw = &s_r[wave][PAD + c0];
  rw[0] = q0; rw[1] = q1; rw[2] = q2; rw[3] = q3;
  rw[4] = q4; rw[5] = q5; rw[6] = q6; rw[7] = q7;

  // totals fill: rows 388..515 (row[388] == P[255] == total, same bits)
  const f32x4 t4 = {total, total, total, total};
  *(f32x4*)(&s_r[wave][388 + lane * 4]) = t4;

  __syncthreads();

  // ---- window sums via two aligned vector gathers -------------------------
  const f32x4 lo0 = *(const f32x4*)(&s_r[wave][c0 + 4]);
  const f32x4 lo1 = *(const f32x4*)(&s_r[wave][c0 + 8]);
  const f32x4 up0 = *(const f32x4*)(&s_r[wave][c0 + 260]);
  const f32x4 up1 = *(const f32x4*)(&s_r[wave][c0 + 264]);

  f32x4 o0, o1;
  o0.x = a.x * rsqrtf(up0.x - lo0.x);
  o0.y = a.y * rsqrtf(up0.y - lo0.y);
  o0.z = a.z * rsqrtf(up0.z - lo0.z);
  o0.w = a.w * rsqrtf(up0.w - lo0.w);
  o1.x = b.x * rsqrtf(up1.x - lo1.x);
  o1.y = b.y * rsqrtf(up1.y - lo1.y);
  o1.z = b.z * rsqrtf(up1.z - lo1.z);
  o1.w = b.w * rsqrtf(up1.w - lo1.w);

  // ---- streaming (non-temporal) output stores -----------------------------
  float* __restrict__ dst = out + elt0;
  __builtin_nontemporal_store(o0, (f32x4*)(dst + c0));
  __builtin_nontemporal_store(o1, (f32x4*)(dst + c0 + 4));
}

extern "C" void kernel_launch(void* const* d_in, const int* in_sizes, int n_in,
                              void* d_out, int out_size, void* d_ws, size_t ws_size,
                              hipStream_t stream) {
  (void)n_in; (void)d_ws; (void)ws_size; (void)out_size;
  const float* x = (const float*)d_in[0];
  float* out = (float*)d_out;
  const int npix = in_sizes[0] / CH;  // 32*56*56 = 100352
  const int blocks = (npix + WPB - 1) / WPB;
  lrn_kernel<<<blocks, THREADS, 0, stream>>>(x, out, npix);
}